// AGNOCrossAttention_46428596470306
// MI455X (gfx1250) — compile-verified
//
#include <hip/hip_runtime.h>
#include <hip/hip_bf16.h>

// AGNO cross-attention for MI455X (gfx1250, wave32).
//
// Phase plan:
//   1) wmma_gemm_f32: C = A @ B (+bias) using V_WMMA_F32_16X16X4_F32,
//      one wave per 16x16 C tile. Used for Qf/Kf/Vf/Gf projections and the
//      final Hout @ Wout + b_out.
//   2) edge_attn: one wave32 per query. q_idx is sorted, so each query's
//      edges are a contiguous range found by binary search. Segmented
//      softmax + weighted V+G accumulation done in registers with
//      shfl_xor reductions per head (lane l owns row elements [4l,4l+4),
//      head = l/4). No atomics -> deterministic.
//
// Workspace layout (floats): Qf[NQ*D] Kf[NS*D] Vf[NS*D] Gf[NQ*D] Hout[NQ*D]
//   = (2*NQ + 2*NS + NQ) * 128 floats ~= 176 MB. Fits standard harness ws.

#define D_DIM 128
#define H_DIM 8

typedef __attribute__((ext_vector_type(2))) float v2f;
typedef __attribute__((ext_vector_type(8))) float v8f;

// ---------------------------------------------------------------------------
// WMMA f32 GEMM: C[M,N] = A[M,K] @ B[K,N] (+ bias[N] if bias != nullptr)
// One wave per 16x16 tile of C. K must be a multiple of 4; M,N multiples of 16.
//
// ISA layouts (CDNA5 7.12.2), f32 16x16x4:
//   A (16x4):  lane l, vgpr r -> A[l&15][r + 2*(l>>4)]
//   B (4x16):  lane l, vgpr r -> B[r + 2*(l>>4)][l&15]
//   C (16x16): lane l, vgpr r -> C[r + 8*(l>>4)][l&15]
// ---------------------------------------------------------------------------
__global__ void wmma_gemm_f32(const float* __restrict__ A,
                              const float* __restrict__ B,
                              float* __restrict__ C,
                              const float* __restrict__ bias,
                              int M, int K, int N) {
  const int wavesPerBlock = blockDim.x >> 5;
  const int wave = blockIdx.x * wavesPerBlock + (threadIdx.x >> 5);
  const int lane = threadIdx.x & 31;
  const int tilesN = N >> 4;
  const int tilesM = M >> 4;
  const int tm = wave / tilesN;
  const int tn = wave % tilesN;
  if (tm >= tilesM) return;           // wave-uniform: EXEC stays all-ones

  const int hi = lane >> 4;           // 0 or 1 (half-wave)
  const int lo = lane & 15;

  v8f c = {};
  const float* arow = A + (size_t)(tm * 16 + lo) * K;
  const float* bcol = B + tn * 16 + lo;

  for (int k0 = 0; k0 < K; k0 += 4) {
    const int ka = k0 + 2 * hi;
    v2f a, b;
    a.x = arow[ka];
    a.y = arow[ka + 1];
    b.x = bcol[(size_t)ka * N];
    b.y = bcol[(size_t)(ka + 1) * N];
    // 8 args: (neg_a, A, neg_b, B, c_mod, C, reuse_a, reuse_b)
    c = __builtin_amdgcn_wmma_f32_16x16x4_f32(false, a, false, b,
                                              (short)0, c, false, false);
  }

  const float bv = bias ? bias[tn * 16 + lo] : 0.0f;
  float* crow = C + (size_t)(tm * 16) * N + tn * 16 + lo;
#pragma unroll
  for (int r = 0; r < 8; ++r) {
    const int m = r + 8 * hi;
    crow[(size_t)m * N] = c[r] + bv;
  }
}

// ---------------------------------------------------------------------------
// Edge attention: one wave32 per query q.
//   pass 1: max over this query's edge scores (seeded with 0.0, matching
//           reference maximum(segment_max, 0))
//   pass 2: recompute score (K rows are L2-resident), accumulate
//           sum(exp) and sum(exp * (V+G)), then normalize.
// ---------------------------------------------------------------------------
__global__ void edge_attn(const float* __restrict__ Qf,
                          const float* __restrict__ Kf,
                          const float* __restrict__ Vf,
                          const float* __restrict__ Gf,
                          const int* __restrict__ q_idx,
                          const int* __restrict__ s_idx,
                          const float* __restrict__ log_tau,
                          float* __restrict__ Hout,
                          int E, int NQ) {
  const int wavesPerBlock = blockDim.x >> 5;
  const int q = blockIdx.x * wavesPerBlock + (threadIdx.x >> 5);
  if (q >= NQ) return;                // wave-uniform exit
  const int lane = threadIdx.x & 31;

  // lower_bound over sorted q_idx for q and q+1 (uniform across the wave)
  int start, end;
  {
    int lo = 0, hi = E;
    while (lo < hi) { int mid = (lo + hi) >> 1; if (q_idx[mid] < q)     lo = mid + 1; else hi = mid; }
    start = lo;
    hi = E;
    while (lo < hi) { int mid = (lo + hi) >> 1; if (q_idx[mid] < q + 1) lo = mid + 1; else hi = mid; }
    end = lo;
  }

  const float4 qv = ((const float4*)(Qf + (size_t)q * D_DIM))[lane];
  const float4 gv = ((const float4*)(Gf + (size_t)q * D_DIM))[lane];
  // scores = dot / (sqrt(DH) * tau) = dot * 0.25 * exp(-log_tau)
  const float scale = 0.25f * __expf(-log_tau[0]);

  // Pass 1: per-head running max (each lane in a 4-lane head group holds it)
  float mx = 0.0f;
  for (int e = start; e < end; ++e) {
    const int s = s_idx[e];
    const float4 kv = ((const float4*)(Kf + (size_t)s * D_DIM))[lane];
    float p = qv.x * kv.x + qv.y * kv.y + qv.z * kv.z + qv.w * kv.w;
    p += __shfl_xor(p, 1, 32);
    p += __shfl_xor(p, 2, 32);        // lanes 4h..4h+3 now all hold head-h dot
    mx = fmaxf(mx, p * scale);
  }

  // Pass 2: sum of exp and weighted accumulation of (V + G)
  float ssum = 0.0f;
  float4 acc = make_float4(0.f, 0.f, 0.f, 0.f);
  for (int e = start; e < end; ++e) {
    const int s = s_idx[e];
    const float4 kv = ((const float4*)(Kf + (size_t)s * D_DIM))[lane];
    float p = qv.x * kv.x + qv.y * kv.y + qv.z * kv.z + qv.w * kv.w;
    p += __shfl_xor(p, 1, 32);
    p += __shfl_xor(p, 2, 32);
    const float w = __expf(p * scale - mx);
    ssum += w;
    const float4 vv = ((const float4*)(Vf + (size_t)s * D_DIM))[lane];
    acc.x += w * (vv.x + gv.x);
    acc.y += w * (vv.y + gv.y);
    acc.z += w * (vv.z + gv.z);
    acc.w += w * (vv.w + gv.w);
  }

  const float inv = 1.0f / fmaxf(ssum, 1e-8f);
  float4 outv = make_float4(acc.x * inv, acc.y * inv, acc.z * inv, acc.w * inv);
  ((float4*)(Hout + (size_t)q * D_DIM))[lane] = outv;
}

// ---------------------------------------------------------------------------
extern "C" void kernel_launch(void* const* d_in, const int* in_sizes, int n_in,
                              void* d_out, int out_size, void* d_ws, size_t ws_size,
                              hipStream_t stream) {
  const float* query   = (const float*)d_in[0];   // [NQ, D]
  const float* support = (const float*)d_in[1];   // [NS, D]
  const float* geo     = (const float*)d_in[2];   // [NQ, G]
  const float* Wq      = (const float*)d_in[3];   // [D, D]
  const float* Wk      = (const float*)d_in[4];
  const float* Wv      = (const float*)d_in[5];
  const float* Wg      = (const float*)d_in[6];   // [G, D]
  const float* Wout    = (const float*)d_in[7];   // [D, D]
  const float* b_out   = (const float*)d_in[8];   // [D]
  const float* log_tau = (const float*)d_in[9];   // [1]
  const int*   q_idx   = (const int*)d_in[10];    // [E] sorted
  const int*   s_idx   = (const int*)d_in[11];    // [E]

  const int NQ = in_sizes[0] / D_DIM;
  const int NS = in_sizes[1] / D_DIM;
  const int G  = in_sizes[6] / D_DIM;             // Wg is [G, D]
  const int E  = in_sizes[10];

  float* ws   = (float*)d_ws;
  float* Qf   = ws;                               // NQ*D
  float* Kf   = Qf + (size_t)NQ * D_DIM;          // NS*D
  float* Vf   = Kf + (size_t)NS * D_DIM;          // NS*D
  float* Gf   = Vf + (size_t)NS * D_DIM;          // NQ*D
  float* Hout = Gf + (size_t)NQ * D_DIM;          // NQ*D

  const int block = 256;                          // 8 waves per block
  const int wpb = block / 32;

  auto gemmBlocks = [&](int M, int N) {
    int tiles = (M / 16) * (N / 16);
    return (tiles + wpb - 1) / wpb;
  };

  // Projections
  wmma_gemm_f32<<<gemmBlocks(NQ, D_DIM), block, 0, stream>>>(query,   Wq, Qf, nullptr, NQ, D_DIM, D_DIM);
  wmma_gemm_f32<<<gemmBlocks(NS, D_DIM), block, 0, stream>>>(support, Wk, Kf, nullptr, NS, D_DIM, D_DIM);
  wmma_gemm_f32<<<gemmBlocks(NS, D_DIM), block, 0, stream>>>(support, Wv, Vf, nullptr, NS, D_DIM, D_DIM);
  wmma_gemm_f32<<<gemmBlocks(NQ, D_DIM), block, 0, stream>>>(geo,     Wg, Gf, nullptr, NQ, G,     D_DIM);

  // Per-query segmented attention (one wave per query, no atomics)
  edge_attn<<<(NQ + wpb - 1) / wpb, block, 0, stream>>>(Qf, Kf, Vf, Gf, q_idx, s_idx,
                                                        log_tau, Hout, E, NQ);

  // Final projection with bias
  wmma_gemm_f32<<<gemmBlocks(NQ, D_DIM), block, 0, stream>>>(Hout, Wout, (float*)d_out,
                                                             b_out, NQ, D_DIM, D_DIM);
}